// ReadingOrderGNN_15693810499653
// MI455X (gfx1250) — compile-verified
//
#include <hip/hip_runtime.h>

// ---------------------------------------------------------------------------
// GCN forward for MI455X (gfx1250, wave32).
//   - fp32 end-to-end (matches reference precision)
//   - dense 64x64 GEMMs via V_WMMA_F32_16X16X4_F32 (64 wmma per wave-tile)
//   - edge propagation via wave-per-edge gather + global_atomic_add_f32
// ---------------------------------------------------------------------------

typedef __attribute__((ext_vector_type(2))) float v2f;
typedef __attribute__((ext_vector_type(8))) float v8f;

#define HID 64

// ---------------------------- degree / norm --------------------------------

__global__ void deg_init_kernel(float* __restrict__ deg, int N) {
    int i = blockIdx.x * blockDim.x + threadIdx.x;
    if (i < N) deg[i] = 1.0f;   // self-loop contributes 1 to every target degree
}

__global__ void deg_count_kernel(const long long* __restrict__ col,
                                 float* __restrict__ deg, int E) {
    int e = blockIdx.x * blockDim.x + threadIdx.x;
    if (e < E) unsafeAtomicAdd(&deg[(int)col[e]], 1.0f);
}

__global__ void dinv_kernel(float* __restrict__ deg, int N) {
    int i = blockIdx.x * blockDim.x + threadIdx.x;
    if (i < N) deg[i] = __frsqrt_rn(deg[i]);   // deg >= 1 always
}

// ------------------------------ layer 1 (K=2) ------------------------------

__global__ void lin1_kernel(const float* __restrict__ x,
                            const float* __restrict__ W1,
                            float* __restrict__ t, int N) {
    int idx = blockIdx.x * blockDim.x + threadIdx.x;   // N*64 threads
    if (idx >= N * HID) return;
    int i = idx >> 6;
    int c = idx & 63;
    t[idx] = fmaf(x[2 * i], W1[c], x[2 * i + 1] * W1[HID + c]);
}

// ------------------------- propagation (A+I norm) --------------------------

// out[i][:] = dinv[i]^2 * t[i][:]   (self-loop term; also initializes out)
__global__ void selfloop_kernel(const float* __restrict__ t,
                                const float* __restrict__ dinv,
                                float* __restrict__ outp, int N) {
    int idx = blockIdx.x * blockDim.x + threadIdx.x;
    if (idx >= N * HID) return;
    float d = dinv[idx >> 6];
    outp[idx] = d * d * t[idx];
}

// one wave32 per edge: out[col] += dinv[row]*dinv[col] * t[row][:]
__global__ void __launch_bounds__(256)
scatter_kernel(const float* __restrict__ t,
               const float* __restrict__ dinv,
               const long long* __restrict__ row,
               const long long* __restrict__ col,
               float* __restrict__ outp, int E) {
    int e = blockIdx.x * (blockDim.x >> 5) + (threadIdx.x >> 5);
    if (e >= E) return;                       // uniform per wave
    int lane = threadIdx.x & 31;
    int r = (int)row[e];                      // wave-uniform -> scalar loads
    int c = (int)col[e];
    float norm = dinv[r] * dinv[c];
    float v0 = t[r * HID + lane];
    float v1 = t[r * HID + lane + 32];
    unsafeAtomicAdd(&outp[c * HID + lane],      norm * v0);
    unsafeAtomicAdd(&outp[c * HID + lane + 32], norm * v1);
}

// ----------------- weight pre-shuffle into B-fragment order ----------------
// B frag (4x16 f32): vgpr0 = W[k][n] with k = 4*kt + 2*(lane>>4), n = 16*nt+(lane&15)
//                    vgpr1 = W[k+1][n]
__global__ void wprep_kernel(const float* __restrict__ W,
                             float* __restrict__ Wb) {
    int tid = blockIdx.x * blockDim.x + threadIdx.x;   // 2048 threads
    if (tid >= 64 * 32) return;
    int frag = tid >> 5;            // 0..63  = kt*4 + nt
    int lane = tid & 31;
    int kt = frag >> 2;
    int nt = frag & 3;
    int k = kt * 4 + ((lane >> 4) << 1);
    int n = nt * 16 + (lane & 15);
    Wb[2 * tid]     = W[k * HID + n];
    Wb[2 * tid + 1] = W[(k + 1) * HID + n];
}

// --------------------------- WMMA fp32 GEMM --------------------------------
// C[N,64] = act(A + bias_prev) @ W, act = relu or identity (fused epilogue of
// the previous GCN conv). 128 threads = 4 waves; each wave: 16 rows x 64 cols.
// DO_RELU is a compile-time template arg so the epilogue lowers to a bare
// v_max (relu) or nothing, with no per-element cndmask selects.
template <bool DO_RELU>
__global__ void __launch_bounds__(128)
gemm_wmma_f32_kernel(const float* __restrict__ A,
                     const float* __restrict__ bias,   // [64] prev-layer bias
                     const float* __restrict__ Wb,     // [64 frags][32] v2f
                     float* __restrict__ C, int N) {
    const int lane  = threadIdx.x & 31;
    const int wave  = threadIdx.x >> 5;
    const int row0  = blockIdx.x * 64 + wave * 16;
    const int mrow  = row0 + (lane & 15);
    const int m     = (mrow < N) ? mrow : (N - 1);   // clamp: EXEC must stay all-1s
    const int khalf = (lane >> 4) << 1;              // 0 or 2

    const v2f* __restrict__ Wb2 = (const v2f*)Wb;

    v8f c0 = {}, c1 = {}, c2 = {}, c3 = {};

    #pragma unroll
    for (int kt = 0; kt < 16; ++kt) {
        const int k = kt * 4 + khalf;
        v2f a;
        a.x = A[m * HID + k]     + bias[k];
        a.y = A[m * HID + k + 1] + bias[k + 1];
        if (DO_RELU) {
            a.x = fmaxf(a.x, 0.0f);
            a.y = fmaxf(a.y, 0.0f);
        }
        v2f b0 = Wb2[(kt * 4 + 0) * 32 + lane];
        v2f b1 = Wb2[(kt * 4 + 1) * 32 + lane];
        v2f b2 = Wb2[(kt * 4 + 2) * 32 + lane];
        v2f b3 = Wb2[(kt * 4 + 3) * 32 + lane];
        c0 = __builtin_amdgcn_wmma_f32_16x16x4_f32(false, a, false, b0, (short)0, c0, false, false);
        c1 = __builtin_amdgcn_wmma_f32_16x16x4_f32(false, a, false, b1, (short)0, c1, false, false);
        c2 = __builtin_amdgcn_wmma_f32_16x16x4_f32(false, a, false, b2, (short)0, c2, false, false);
        c3 = __builtin_amdgcn_wmma_f32_16x16x4_f32(false, a, false, b3, (short)0, c3, false, false);
    }

    // C/D layout: vgpr r -> row (r + 8*(lane>>4)), col (lane&15) within 16-tile
    const int crow0 = row0 + ((lane >> 4) << 3);
    const int ccol  = lane & 15;
    #pragma unroll
    for (int r = 0; r < 8; ++r) {
        int rr = crow0 + r;
        if (rr < N) {
            float* dst = C + rr * HID + ccol;
            dst[0]  = c0[r];
            dst[16] = c1[r];
            dst[32] = c2[r];
            dst[48] = c3[r];
        }
    }
}

// ---------------- final head: out = relu(g + b4) @ W5 + b5 ----------------

__global__ void __launch_bounds__(256)
head_kernel(const float* __restrict__ g,
            const float* __restrict__ b4,
            const float* __restrict__ W5,
            const float* __restrict__ b5,
            float* __restrict__ out, int N) {
    int i = blockIdx.x * (blockDim.x >> 5) + (threadIdx.x >> 5);  // node per wave
    if (i >= N) return;
    int lane = threadIdx.x & 31;
    float p = fmaxf(g[i * HID + lane]      + b4[lane],      0.0f) * W5[lane]
            + fmaxf(g[i * HID + lane + 32] + b4[lane + 32], 0.0f) * W5[lane + 32];
    #pragma unroll
    for (int off = 16; off >= 1; off >>= 1) p += __shfl_xor(p, off, 32);
    if (lane == 0) out[i] = p + b5[0];
}

// ---------------------------------------------------------------------------

static inline int cdiv(int a, int b) { return (a + b - 1) / b; }

extern "C" void kernel_launch(void* const* d_in, const int* in_sizes, int n_in,
                              void* d_out, int out_size, void* d_ws, size_t ws_size,
                              hipStream_t stream) {
    const float*     x     = (const float*)d_in[0];
    const long long* edges = (const long long*)d_in[1];   // int64 [2, E]
    // d_in[2] = batch (unused)
    const float* W1 = (const float*)d_in[3];
    const float* b1 = (const float*)d_in[4];
    const float* W2 = (const float*)d_in[5];
    const float* b2 = (const float*)d_in[6];
    const float* W3 = (const float*)d_in[7];
    const float* b3 = (const float*)d_in[8];
    const float* W4 = (const float*)d_in[9];
    const float* b4 = (const float*)d_in[10];
    const float* W5 = (const float*)d_in[11];
    const float* b5 = (const float*)d_in[12];

    const int N = in_sizes[0] / 2;        // x is [N, 2]
    const int E = in_sizes[1] / 2;        // edge_index is [2, E]
    const long long* erow = edges;        // sources
    const long long* ecol = edges + E;    // targets

    // ---- workspace carve-out (floats, 256B aligned chunks) ----
    float* ws = (float*)d_ws;
    size_t off = 0;
    auto alloc = [&](size_t nfloats) {
        float* p = ws + off;
        off += (nfloats + 63) & ~(size_t)63;
        return p;
    };
    float* dinv = alloc((size_t)N);            // degree -> rsqrt in place
    float* bufA = alloc((size_t)N * HID);
    float* bufB = alloc((size_t)N * HID);
    float* Wb2  = alloc(64 * 32 * 2);
    float* Wb3  = alloc(64 * 32 * 2);
    float* Wb4  = alloc(64 * 32 * 2);
    (void)ws_size; (void)n_in; (void)out_size;

    const int NT = N * HID;

    // degrees + symmetric norm
    deg_init_kernel<<<cdiv(N, 256), 256, 0, stream>>>(dinv, N);
    deg_count_kernel<<<cdiv(E, 256), 256, 0, stream>>>(ecol, dinv, E);
    dinv_kernel<<<cdiv(N, 256), 256, 0, stream>>>(dinv, N);

    // weight fragment pre-shuffle (B-matrix layout for WMMA f32 16x16x4)
    wprep_kernel<<<8, 256, 0, stream>>>(W2, Wb2);
    wprep_kernel<<<8, 256, 0, stream>>>(W3, Wb3);
    wprep_kernel<<<8, 256, 0, stream>>>(W4, Wb4);

    // layer 1: t = x @ W1   (K = 2, plain VALU)
    lin1_kernel<<<cdiv(NT, 256), 256, 0, stream>>>(x, W1, bufA, N);

    // conv1 propagate: bufB = D^-1/2 (A+I) D^-1/2 bufA
    selfloop_kernel<<<cdiv(NT, 256), 256, 0, stream>>>(bufA, dinv, bufB, N);
    scatter_kernel<<<cdiv(E, 8), 256, 0, stream>>>(bufA, dinv, erow, ecol, bufB, E);

    // conv2 matmul: bufA = relu(bufB + b1) @ W2
    gemm_wmma_f32_kernel<true><<<cdiv(N, 64), 128, 0, stream>>>(bufB, b1, Wb2, bufA, N);
    selfloop_kernel<<<cdiv(NT, 256), 256, 0, stream>>>(bufA, dinv, bufB, N);
    scatter_kernel<<<cdiv(E, 8), 256, 0, stream>>>(bufA, dinv, erow, ecol, bufB, E);

    // conv3 matmul: bufA = relu(bufB + b2) @ W3
    gemm_wmma_f32_kernel<true><<<cdiv(N, 64), 128, 0, stream>>>(bufB, b2, Wb3, bufA, N);
    selfloop_kernel<<<cdiv(NT, 256), 256, 0, stream>>>(bufA, dinv, bufB, N);
    scatter_kernel<<<cdiv(E, 8), 256, 0, stream>>>(bufA, dinv, erow, ecol, bufB, E);

    // dense: g = (bufB + b3) @ W4   (no relu on input; relu+b4 fused in head)
    gemm_wmma_f32_kernel<false><<<cdiv(N, 64), 128, 0, stream>>>(bufB, b3, Wb4, bufA, N);

    // head: out = relu(g + b4) @ W5 + b5
    head_kernel<<<cdiv(N, 8), 256, 0, stream>>>(bufA, b4, W5, b5, (float*)d_out, N);
}